// Refine_module_31104153157905
// MI455X (gfx1250) — compile-verified
//
#include <hip/hip_runtime.h>
#include <math.h>

// ============================================================================
// Refine_module forward for MI455X (gfx1250, wave32).
// All dense layers over row-major activations use V_WMMA_F32_16X16X4_F32
// (exact fp32 matrix math). Irregular ops (LN, softmax, top-k, gathers,
// frame update) are VALU kernels.
//
// Assumed d_in flattening (recursive dict-insertion order, dense = w then b,
// LN tuple = gamma then beta):
//   0 node (1,384,64)  1 edge (1,384,384,128)  2 seq1hot (1,384,21)  3 idx (1,384) i32
//   4.. params leaves (see P_* constants below), 98 tensors total.
// ============================================================================

typedef float v2f __attribute__((ext_vector_type(2)));
typedef float v8f __attribute__((ext_vector_type(8)));

#define LSEQ  384
#define NPAIR (LSEQ * LSEQ)   // 147456
#define NHEAD 4
#define TOPK  64
#define LK    (LSEQ * TOPK)   // 24576

__device__ __forceinline__ float siluf(float x) { return x / (1.f + __expf(-x)); }

// ---------------------------------------------------------------------------
// WMMA f32 GEMM:  C[M,N] = act(A[M,K] @ W[K,N] + bias)
// One wave per 16x32 output tile (two 16x16 accumulators sharing one A frag).
// Main K loop is branch-free, software-pipelined, and uses strength-reduced
// running row pointers (no 64-bit multiplies in the loop; second column tile
// reached via +64B immediate offset). Ragged K tail uses clamped addresses +
// value masks (no exec divergence).
// Requires: M % 16 == 0, N % 32 == 0, K >= 4 (true for every call here).
// ---------------------------------------------------------------------------
__global__ void k_gemm_wmma(const float* __restrict__ A, int lda,
                            const float* __restrict__ W, const float* __restrict__ bias,
                            float* __restrict__ C, int ldc,
                            int M, int N, int K, int act)
{
    const int lane = threadIdx.x & 31;
    const int wave = threadIdx.x >> 5;
    const int wpb  = blockDim.x >> 5;
    const long tilesN = N >> 5;
    const long tilesM = M >> 4;
    long tile = (long)blockIdx.x * wpb + wave;
    if (tile >= tilesM * tilesN) return;            // wave-uniform: EXEC all-1s at WMMA
    const int tm   = (int)(tile / tilesN) << 4;
    const int tn   = (int)(tile % tilesN) << 5;
    const int r15  = lane & 15;
    const int hi   = lane >> 4;                     // 0: K={k0,k0+1}, 1: K={k0+2,k0+3}
    const float* Ap   = A + (long)(tm + r15) * lda;
    const int col0 = tn + r15;

    v8f acc0 = {}, acc1 = {};
    const int Kmain = K & ~3;
    const long stepW = 4L * N;

    // running pointers: pa -> A[row, k], pwA -> W[k, col0], pwB -> W[k+1, col0]
    const float* pa  = Ap + 2 * hi;
    const float* pwA = W + (long)(2 * hi) * N + col0;
    const float* pwB = pwA + N;

    // prologue: fragments for k-step 0
    v2f a0, b0, b1;
    a0.x = pa[0];   a0.y = pa[1];
    b0.x = pwA[0];  b0.y = pwB[0];
    b1.x = pwA[16]; b1.y = pwB[16];                 // +64B immediate offset
    pa += 4; pwA += stepW; pwB += stepW;

    for (int k0 = 4; k0 < Kmain; k0 += 4) {
        v2f a1, c0, c1;                             // prefetch next step before WMMA
        a1.x = pa[0];   a1.y = pa[1];
        c0.x = pwA[0];  c0.y = pwB[0];
        c1.x = pwA[16]; c1.y = pwB[16];
        pa += 4; pwA += stepW; pwB += stepW;
        acc0 = __builtin_amdgcn_wmma_f32_16x16x4_f32(false, a0, false, b0, (short)0, acc0, false, false);
        acc1 = __builtin_amdgcn_wmma_f32_16x16x4_f32(false, a0, false, b1, (short)0, acc1, false, false);
        a0 = a1; b0 = c0; b1 = c1;
    }
    acc0 = __builtin_amdgcn_wmma_f32_16x16x4_f32(false, a0, false, b0, (short)0, acc0, false, false);
    acc1 = __builtin_amdgcn_wmma_f32_16x16x4_f32(false, a0, false, b1, (short)0, acc1, false, false);

    if (Kmain < K) {                    // 1..3 leftover K values
        const int kt  = Kmain + 2 * hi;
        const int k0c = (kt     < K) ? kt     : (K - 1);   // clamped (always in-range)
        const int k1c = (kt + 1 < K) ? kt + 1 : (K - 1);
        const float m0 = (kt     < K) ? 1.f : 0.f;         // value masks -> v_cndmask
        const float m1 = (kt + 1 < K) ? 1.f : 0.f;
        v2f a, b, c;
        a.x = Ap[k0c] * m0;                  a.y = Ap[k1c] * m1;
        b.x = W[(long)k0c * N + col0];       b.y = W[(long)k1c * N + col0];
        c.x = W[(long)k0c * N + col0 + 16];  c.y = W[(long)k1c * N + col0 + 16];
        acc0 = __builtin_amdgcn_wmma_f32_16x16x4_f32(false, a, false, b, (short)0, acc0, false, false);
        acc1 = __builtin_amdgcn_wmma_f32_16x16x4_f32(false, a, false, c, (short)0, acc1, false, false);
    }

    const float bv0 = bias ? bias[col0] : 0.f;
    const float bv1 = bias ? bias[col0 + 16] : 0.f;
#pragma unroll
    for (int r = 0; r < 8; ++r) {
        long orow = (long)(tm + r + 8 * hi);
        float v0 = acc0[r] + bv0;
        float v1 = acc1[r] + bv1;
        if (act == 1) { v0 = siluf(v0); v1 = siluf(v1); }
        C[orow * ldc + col0]      = v0;
        C[orow * ldc + col0 + 16] = v1;
    }
}

// ---------------------------------------------------------------------------
// LayerNorm over last dim C: one wave32 per row (in-place safe).
// ---------------------------------------------------------------------------
__global__ void k_ln(const float* __restrict__ x, int ldx,
                     const float* __restrict__ g, const float* __restrict__ b,
                     float* __restrict__ y, int ldy, long rows, int C)
{
    const int wpb = blockDim.x >> 5;
    long row = (long)blockIdx.x * wpb + (threadIdx.x >> 5);
    if (row >= rows) return;
    const int lane = threadIdx.x & 31;
    const float* xp = x + row * ldx;
    float s1 = 0.f, s2 = 0.f;
    for (int c = lane; c < C; c += 32) { float v = xp[c]; s1 += v; s2 += v * v; }
    for (int m = 16; m > 0; m >>= 1) { s1 += __shfl_xor(s1, m, 32); s2 += __shfl_xor(s2, m, 32); }
    float mean = s1 / C;
    float var  = s2 / C - mean * mean;
    float inv  = rsqrtf(var + 1e-5f);
    float* yp = y + row * ldy;
    for (int c = lane; c < C; c += 32) yp[c] = (xp[c] - mean) * inv * g[c] + b[c];
}

// Small linear: one thread per output element; act 0=none,1=silu,2=sigmoid.
__global__ void k_lin(const float* __restrict__ A, int lda,
                      const float* __restrict__ W, const float* __restrict__ b,
                      float* __restrict__ C, int ldc, long M, int N, int K, int act)
{
    long t = (long)blockIdx.x * blockDim.x + threadIdx.x;
    if (t >= M * N) return;
    long m = t / N; int n = (int)(t % N);
    float acc = b ? b[n] : 0.f;
    for (int k = 0; k < K; ++k) acc += A[m * lda + k] * W[(long)k * N + n];
    if (act == 1) acc = siluf(acc);
    else if (act == 2) acc = 1.f / (1.f + __expf(-acc));
    C[m * ldc + n] = acc;
}

// Concat up to 3 row-major sources into y (ld = ca+cb+cc).
__global__ void k_concat3(const float* a, int ca, const float* b, int cb,
                          const float* c, int cc, float* y, long rows)
{
    const int tot = ca + cb + cc;
    long t = (long)blockIdx.x * blockDim.x + threadIdx.x;
    if (t >= rows * tot) return;
    long r = t / tot; int col = (int)(t % tot);
    float v;
    if (col < ca)            v = a[r * ca + col];
    else if (col < ca + cb)  v = b[r * cb + (col - ca)];
    else                     v = c[r * cc + (col - ca - cb)];
    y[r * tot + col] = v;
}

// seqsep + bonded columns of the 130-wide regen edge features.
__global__ void k_edge_extra(const int* __restrict__ idx, float* __restrict__ y,
                             int ld, int col0)
{
    long t = (long)blockIdx.x * blockDim.x + threadIdx.x;
    if (t >= NPAIR) return;
    int i = (int)(t / LSEQ), j = (int)(t % LSEQ);
    float sep = (float)(idx[j] - idx[i]);
    float a = fabsf(sep);
    float sgn = (sep > 0.f) ? 1.f : ((sep < 0.f) ? -1.f : 0.f);
    float ss = sgn * fminf(fmaxf(logf(a + 1.f), 0.f), 5.5f);
    float bn = (a <= 1.f) ? sep : 0.f;
    y[t * ld + col0]     = ss;
    y[t * ld + col0 + 1] = bn;
}

// ----------------------- UniMP attention (VALU) ----------------------------
// scores[(j*4+h)*L + i] = q[j,h,:].(k[i,h,:]+e[i,j,h,:])/8, -1e9 on i==j
__global__ void k_scores(const float* __restrict__ q, const float* __restrict__ k,
                         const float* __restrict__ e, float* __restrict__ sc)
{
    long t = (long)blockIdx.x * blockDim.x + threadIdx.x;
    if (t >= (long)LSEQ * NHEAD * LSEQ) return;
    int i = (int)(t % LSEQ);
    long jh = t / LSEQ;
    int h = (int)(jh % NHEAD), j = (int)(jh / NHEAD);
    const float* qp = q + (long)j * 256 + h * 64;
    const float* kp = k + (long)i * 256 + h * 64;
    const float* ep = e + ((long)i * LSEQ + j) * 256 + h * 64;
    float acc = 0.f;
    for (int d = 0; d < 64; ++d) acc += qp[d] * (kp[d] + ep[d]);
    sc[t] = (i == j) ? -1e9f : acc * 0.125f;
}

__global__ void k_softmax(float* __restrict__ s, int len)
{
    float* p = s + (long)blockIdx.x * len;
    __shared__ float red[128];
    float mx = -1e30f;
    for (int i = threadIdx.x; i < len; i += blockDim.x) mx = fmaxf(mx, p[i]);
    red[threadIdx.x] = mx; __syncthreads();
    for (int st = 64; st > 0; st >>= 1) {
        if (threadIdx.x < st) red[threadIdx.x] = fmaxf(red[threadIdx.x], red[threadIdx.x + st]);
        __syncthreads();
    }
    mx = red[0]; __syncthreads();
    float sm = 0.f;
    for (int i = threadIdx.x; i < len; i += blockDim.x) { float e = __expf(p[i] - mx); p[i] = e; sm += e; }
    red[threadIdx.x] = sm; __syncthreads();
    for (int st = 64; st > 0; st >>= 1) {
        if (threadIdx.x < st) red[threadIdx.x] += red[threadIdx.x + st];
        __syncthreads();
    }
    float inv = 1.f / red[0];
    for (int i = threadIdx.x; i < len; i += blockDim.x) p[i] *= inv;
}

// out[j,hd] += sum_i alpha[j,i,h]*(v[i,hd]+e[i,j,hd]); block per j, 256 thr.
__global__ void k_attn_out(const float* __restrict__ al, const float* __restrict__ v,
                           const float* __restrict__ e, float* __restrict__ out)
{
    int j = blockIdx.x, hd = threadIdx.x, h = hd >> 6;
    float acc = 0.f;
    for (int i = 0; i < LSEQ; ++i) {
        float a = al[((long)j * NHEAD + h) * LSEQ + i];
        acc += a * (v[(long)i * 256 + hd] + e[((long)i * LSEQ + j) * 256 + hd]);
    }
    out[(long)j * 256 + hd] += acc;
}

__global__ void k_elu_res(float* __restrict__ x, const float* __restrict__ d, long n)
{
    long t = (long)blockIdx.x * blockDim.x + threadIdx.x;
    if (t >= n) return;
    float v = x[t] + d[t];
    x[t] = (v > 0.f) ? v : (__expf(v) - 1.f);
}

// ------------------------------ refine helpers -----------------------------
__global__ void k_ca(const float* __restrict__ xyz, float* __restrict__ ca)
{
    int t = blockIdx.x * blockDim.x + threadIdx.x;
    if (t >= LSEQ * 3) return;
    ca[t] = xyz[(t / 3) * 9 + 3 + (t % 3)];
}

__global__ void k_dist(const float* __restrict__ ca, float* __restrict__ D)
{
    long t = (long)blockIdx.x * blockDim.x + threadIdx.x;
    if (t >= NPAIR) return;
    int i = (int)(t / LSEQ), j = (int)(t % LSEQ);
    float dx = ca[i * 3 + 0] - ca[j * 3 + 0];
    float dy = ca[i * 3 + 1] - ca[j * 3 + 1];
    float dz = ca[i * 3 + 2] - ca[j * 3 + 2];
    D[t] = sqrtf(dx * dx + dy * dy + dz * dz + 1e-8f);
}

// rbf(D) (36) + bonded (1) at columns [32..68] of the 69-wide pair features.
__global__ void k_pair_extra(const float* __restrict__ D, const int* __restrict__ idx,
                             float* __restrict__ y)
{
    long t = (long)blockIdx.x * blockDim.x + threadIdx.x;
    if (t >= NPAIR) return;
    float d = D[t];
    float* row = y + t * 69;
#pragma unroll 4
    for (int r = 0; r < 36; ++r) {
        float mu = 20.f * (float)r / 35.f;
        float z = (d - mu) * (36.f / 20.f);
        row[32 + r] = __expf(-z * z);
    }
    int i = (int)(t / LSEQ), j = (int)(t % LSEQ);
    float sep = (float)(idx[j] - idx[i]);
    row[68] = (fabsf(sep) <= 1.f) ? sep : 0.f;
}

// 64 nearest neighbors per row (ascending D, low-index tie-break like top_k).
__global__ void k_topk(const float* __restrict__ D, int* __restrict__ eidx)
{
    int i = blockIdx.x;
    __shared__ float vals[LSEQ];
    __shared__ float rv[128];
    __shared__ int   ri[128];
    for (int j = threadIdx.x; j < LSEQ; j += blockDim.x)
        vals[j] = D[(long)i * LSEQ + j] + ((j == i) ? 1e6f : 0.f);
    __syncthreads();
    for (int sel = 0; sel < TOPK; ++sel) {
        float bv = 1e30f; int bi = LSEQ;
        for (int j = threadIdx.x; j < LSEQ; j += blockDim.x) {
            float v = vals[j];
            if (v < bv || (v == bv && j < bi)) { bv = v; bi = j; }
        }
        rv[threadIdx.x] = bv; ri[threadIdx.x] = bi; __syncthreads();
        for (int st = 64; st > 0; st >>= 1) {
            if (threadIdx.x < st) {
                float ov = rv[threadIdx.x + st]; int oi = ri[threadIdx.x + st];
                if (ov < rv[threadIdx.x] || (ov == rv[threadIdx.x] && oi < ri[threadIdx.x])) {
                    rv[threadIdx.x] = ov; ri[threadIdx.x] = oi;
                }
            }
            __syncthreads();
        }
        if (threadIdx.x == 0) { eidx[i * TOPK + sel] = ri[0]; vals[ri[0]] = 1e30f; }
        __syncthreads();
    }
}

// feat[(i,k), 132] = [h_i(32) | h_j(32) | w_ij(32) | rbf(dist)(36)]; also dhat, dist.
__global__ void k_feat(const float* __restrict__ h, const float* __restrict__ pair,
                       const float* __restrict__ ca, const int* __restrict__ eidx,
                       float* __restrict__ feat, float* __restrict__ dhat,
                       float* __restrict__ dist)
{
    int t = blockIdx.x * blockDim.x + threadIdx.x;
    if (t >= LK) return;
    int i = t / TOPK;
    int j = eidx[t];
    float dx = ca[j * 3 + 0] - ca[i * 3 + 0];
    float dy = ca[j * 3 + 1] - ca[i * 3 + 1];
    float dz = ca[j * 3 + 2] - ca[i * 3 + 2];
    float dn = sqrtf(dx * dx + dy * dy + dz * dz + 1e-8f);
    dist[t] = dn;
    float inv = 1.f / dn;
    dhat[t * 3 + 0] = dx * inv; dhat[t * 3 + 1] = dy * inv; dhat[t * 3 + 2] = dz * inv;
    float* f = feat + (long)t * 132;
    const float* hi = h + i * 32;
    const float* hj = h + j * 32;
    const float* w  = pair + ((long)i * LSEQ + j) * 32;
    for (int c = 0; c < 32; ++c) { f[c] = hi[c]; f[32 + c] = hj[c]; f[64 + c] = w[c]; }
#pragma unroll 4
    for (int r = 0; r < 36; ++r) {
        float mu = 20.f * (float)r / 35.f;
        float z = (dn - mu) * (36.f / 20.f);
        f[96 + r] = __expf(-z * z);
    }
}

// cat96[i] = [h(32) | sum_k m(64)]
__global__ void k_cat96(const float* __restrict__ h, const float* __restrict__ m,
                        float* __restrict__ y)
{
    int t = blockIdx.x * blockDim.x + threadIdx.x;
    if (t >= LSEQ * 96) return;
    int i = t / 96, c = t % 96;
    float v;
    if (c < 32) v = h[i * 32 + c];
    else {
        v = 0.f; int cc = c - 32;
        for (int k = 0; k < TOPK; ++k) v += m[((long)i * TOPK + k) * 64 + cc];
    }
    y[t] = v;
}

// Frame update: off = s^T dhat + gate @ l1; new backbone coords.
__global__ void k_update(const float* __restrict__ xyz_in, const float* __restrict__ ca,
                         const float* __restrict__ s, const float* __restrict__ dhat,
                         const float* __restrict__ gate, float* __restrict__ xyz_out)
{
    int i = blockIdx.x * blockDim.x + threadIdx.x;
    if (i >= LSEQ) return;
    float off[3][3] = {{0.f}};
    for (int k = 0; k < TOPK; ++k) {
        const float* sp = s    + ((long)i * TOPK + k) * 3;
        const float* dp = dhat + ((long)i * TOPK + k) * 3;
        for (int c = 0; c < 3; ++c)
            for (int x = 0; x < 3; ++x) off[c][x] += sp[c] * dp[x];
    }
    float l1[3][3];
    for (int v = 0; v < 3; ++v)
        for (int x = 0; x < 3; ++x) l1[v][x] = xyz_in[i * 9 + v * 3 + x] - ca[i * 3 + x];
    const float* g = gate + i * 9;
    for (int c = 0; c < 3; ++c)
        for (int x = 0; x < 3; ++x) {
            float a = 0.f;
            for (int v = 0; v < 3; ++v) a += g[c * 3 + v] * l1[v][x];
            off[c][x] += a;
        }
    float can[3];
    for (int x = 0; x < 3; ++x) {
        can[x] = ca[i * 3 + x] + off[1][x];
        xyz_out[i * 9 + 0 + x] = can[x] + off[0][x];
        xyz_out[i * 9 + 3 + x] = can[x];
        xyz_out[i * 9 + 6 + x] = can[x] + off[2][x];
    }
}

__global__ void k_lddt(const float* __restrict__ st, const float* __restrict__ g,
                       const float* __restrict__ b, const float* __restrict__ w,
                       const float* __restrict__ b0, float* __restrict__ out)
{
    int i = blockIdx.x * blockDim.x + threadIdx.x;
    if (i >= LSEQ) return;
    const float* x = st + i * 16;
    float s1 = 0.f, s2 = 0.f;
    for (int c = 0; c < 16; ++c) { s1 += x[c]; s2 += x[c] * x[c]; }
    float mean = s1 / 16.f, var = s2 / 16.f - mean * mean;
    float inv = rsqrtf(var + 1e-5f);
    float acc = b0[0];
    for (int c = 0; c < 16; ++c) acc += ((x[c] - mean) * inv * g[c] + b[c]) * w[c];
    out[i] = 1.f / (1.f + __expf(-acc));
}

__global__ void k_copy(const float* __restrict__ a, float* __restrict__ y, long n)
{
    long t = (long)blockIdx.x * blockDim.x + threadIdx.x;
    if (t < n) y[t] = a[t];
}

// ============================================================================
// Host orchestration
// ============================================================================
static inline int cdiv(long a, long b) { return (int)((a + b - 1) / b); }

extern "C" void kernel_launch(void* const* d_in, const int* in_sizes, int n_in,
                              void* d_out, int out_size, void* d_ws, size_t ws_size,
                              hipStream_t stream)
{
    (void)in_sizes; (void)n_in; (void)out_size; (void)ws_size;
    const float* node    = (const float*)d_in[0];
    const float* edge    = (const float*)d_in[1];
    const float* seq1hot = (const float*)d_in[2];
    const int*   idx     = (const int*)d_in[3];
    float* out = (float*)d_out;

    // --- parameter indices (recursive insertion-order flatten; see header) ---
    const int P_PE_W = 4, P_PE_B = 5;
    const int R_NN_G = 6, R_NN_B = 7, R_NE_G = 8, R_NE_B = 9;
    const int R_EX_W = 10, R_EX_B = 11, R_EXLN_G = 12, R_EXLN_B = 13;
    const int R_EE_W = 14, R_EE_B = 15, R_EELN_G = 16, R_EELN_B = 17;
    const int BLK0 = 18;            // 3 blocks x 14 leaves
    const int R_GX_W = 60, R_GX_B = 61, R_NS_G = 62, R_NS_B = 63, R_GS_W = 64, R_GS_B = 65;
    const int F_NM_G = 66, F_NM_B = 67, F_NP_G = 68, F_NP_B = 69, F_NS_G = 70, F_NS_B = 71;
    const int F_EX_W = 72, F_EX_B = 73, F_NN_G = 74, F_NN_B = 75;
    const int F_E1_W = 76, F_E1_B = 77, F_NE1_G = 78, F_NE1_B = 79;
    const int F_E2_W = 80, F_E2_B = 81, F_NE2_G = 82, F_NE2_B = 83;
    const int F_M1_W = 84, F_M1_B = 85, F_M2_W = 86, F_M2_B = 87;
    const int F_NU_W = 88, F_NU_B = 89, F_VW_W = 90, F_VW_B = 91, F_VM_W = 92, F_VM_B = 93;
    const int P_NSF_G = 94, P_NSF_B = 95, P_LD_W = 96, P_LD_B = 97;
    auto P = [&](int i) -> const float* { return (const float*)d_in[i]; };

    // --- workspace layout (floats) ---
    float* W = (float*)d_ws;
    long o = 0;
    float* EDGE_P = W + o; o += (long)NPAIR * 128;
    float* BIGA   = W + o; o += (long)NPAIR * 132;   // 130-wide edge feats / 69-wide pair2 / 132-wide msg feats
    float* EEMB   = W + o; o += (long)NPAIR * 64;    // regen edge embed; refine pair(32) reuse
    float* EBUF   = W + o; o += (long)NPAIR * 256;   // e projection; refine pair-LN(128) reuse
    float* MBUF   = W + o; o += (long)LK * 64;
    float* MBUF2  = W + o; o += (long)LK * 64;
    float* SBUF   = W + o; o += (long)LK * 3;
    float* DHAT   = W + o; o += (long)LK * 3;
    float* DIST   = W + o; o += (long)LK;
    float* DMAT   = W + o; o += (long)NPAIR;
    float* CA     = W + o; o += LSEQ * 3;
    float* NLN    = W + o; o += LSEQ * 64;
    float* NCAT   = W + o; o += LSEQ * 101;
    float* X      = W + o; o += LSEQ * 64;
    float* T64    = W + o; o += LSEQ * 64;
    float* QB     = W + o; o += LSEQ * 256;
    float* KB     = W + o; o += LSEQ * 256;
    float* VB     = W + o; o += LSEQ * 256;
    float* SC     = W + o; o += (long)LSEQ * NHEAD * LSEQ;
    float* OUTB   = W + o; o += LSEQ * 256;
    float* OLN    = W + o; o += LSEQ * 256;
    float* XYZ_A  = W + o; o += LSEQ * 9;
    float* XYZ_B  = W + o; o += LSEQ * 9;
    float* ST_A   = W + o; o += LSEQ * 16;
    float* ST_B   = W + o; o += LSEQ * 16;
    float* H32    = W + o; o += LSEQ * 32;
    float* ST16   = W + o; o += LSEQ * 16;
    float* GATE   = W + o; o += LSEQ * 9;
    float* CAT96  = W + o; o += LSEQ * 96;
    int*   EIDX   = (int*)(W + o); o += LSEQ * TOPK;

    auto gemm = [&](const float* A, int lda, const float* Wp, const float* bp,
                    float* C, int ldc, int M, int N, int K, int act) {
        long tiles = (long)(M / 16) * (N / 32);          // 16x32 tile per wave
        k_gemm_wmma<<<cdiv(tiles, 8), 256, 0, stream>>>(A, lda, Wp, bp, C, ldc, M, N, K, act);
    };
    auto ln = [&](const float* x, int ldx, const float* g, const float* b,
                  float* y, int ldy, long rows, int C) {
        k_ln<<<cdiv(rows, 8), 256, 0, stream>>>(x, ldx, g, b, y, ldy, rows, C);
    };
    auto lin = [&](const float* A, int lda, const float* Wp, const float* bp,
                   float* C, int ldc, long M, int N, int K, int act) {
        k_lin<<<cdiv(M * N, 256), 256, 0, stream>>>(A, lda, Wp, bp, C, ldc, M, N, K, act);
    };

    // ======================= proj_edge =======================
    gemm(edge, 128, P(P_PE_W), P(P_PE_B), EDGE_P, 128, NPAIR, 128, 128, 0);

    // ======================= regen_net =======================
    ln(node, 64, P(R_NN_G), P(R_NN_B), NLN, 64, LSEQ, 64);
    k_concat3<<<cdiv((long)LSEQ * 85, 256), 256, 0, stream>>>(NLN, 64, seq1hot, 21, nullptr, 0, NCAT, LSEQ);
    gemm(NCAT, 85, P(R_EX_W), P(R_EX_B), T64, 64, LSEQ, 64, 85, 0);
    ln(T64, 64, P(R_EXLN_G), P(R_EXLN_B), X, 64, LSEQ, 64);

    ln(EDGE_P, 128, P(R_NE_G), P(R_NE_B), BIGA, 130, NPAIR, 128);
    k_edge_extra<<<cdiv(NPAIR, 256), 256, 0, stream>>>(idx, BIGA, 130, 128);
    gemm(BIGA, 130, P(R_EE_W), P(R_EE_B), EEMB, 64, NPAIR, 64, 130, 0);
    ln(EEMB, 64, P(R_EELN_G), P(R_EELN_B), EEMB, 64, NPAIR, 64);

    for (int blk = 0; blk < 3; ++blk) {
        int pb = BLK0 + 14 * blk;
        gemm(X, 64, P(pb + 0), P(pb + 1), QB, 256, LSEQ, 256, 64, 0);      // q
        gemm(X, 64, P(pb + 2), P(pb + 3), KB, 256, LSEQ, 256, 64, 0);      // k
        gemm(X, 64, P(pb + 4), P(pb + 5), VB, 256, LSEQ, 256, 64, 0);      // v
        gemm(EEMB, 64, P(pb + 6), P(pb + 7), EBUF, 256, NPAIR, 256, 64, 0);// e (big)
        k_scores<<<cdiv((long)LSEQ * NHEAD * LSEQ, 256), 256, 0, stream>>>(QB, KB, EBUF, SC);
        k_softmax<<<LSEQ * NHEAD, 128, 0, stream>>>(SC, LSEQ);
        gemm(X, 64, P(pb + 8), P(pb + 9), OUTB, 256, LSEQ, 256, 64, 0);    // skip
        k_attn_out<<<LSEQ, 256, 0, stream>>>(SC, VB, EBUF, OUTB);
        ln(OUTB, 256, P(pb + 10), P(pb + 11), OLN, 256, LSEQ, 256);
        gemm(OLN, 256, P(pb + 12), P(pb + 13), T64, 64, LSEQ, 64, 256, 0); // lin
        k_elu_res<<<cdiv((long)LSEQ * 64, 256), 256, 0, stream>>>(X, T64, (long)LSEQ * 64);
    }
    lin(X, 64, P(R_GX_W), P(R_GX_B), XYZ_A, 9, LSEQ, 9, 64, 0);
    ln(X, 64, P(R_NS_G), P(R_NS_B), T64, 64, LSEQ, 64);
    lin(T64, 64, P(R_GS_W), P(R_GS_B), ST_A, 16, LSEQ, 16, 64, 0);

    // ======================= refine_net x2 =======================
    float* xin = XYZ_A; float* sin_ = ST_A; float* xout = XYZ_B; float* sout = ST_B;
    for (int it = 0; it < 2; ++it) {
        ln(node, 64, P(F_NM_G), P(F_NM_B), NLN, 64, LSEQ, 64);
        ln(sin_, 16, P(F_NS_G), P(F_NS_B), ST16, 16, LSEQ, 16);
        k_concat3<<<cdiv((long)LSEQ * 101, 256), 256, 0, stream>>>(NLN, 64, seq1hot, 21, ST16, 16, NCAT, LSEQ);
        gemm(NCAT, 101, P(F_EX_W), P(F_EX_B), H32, 32, LSEQ, 32, 101, 0);
        ln(H32, 32, P(F_NN_G), P(F_NN_B), H32, 32, LSEQ, 32);

        ln(EDGE_P, 128, P(F_NP_G), P(F_NP_B), EBUF, 128, NPAIR, 128);
        gemm(EBUF, 128, P(F_E1_W), P(F_E1_B), BIGA, 69, NPAIR, 32, 128, 0);
        ln(BIGA, 69, P(F_NE1_G), P(F_NE1_B), BIGA, 69, NPAIR, 32);

        k_ca<<<cdiv(LSEQ * 3, 256), 256, 0, stream>>>(xin, CA);
        k_dist<<<cdiv(NPAIR, 256), 256, 0, stream>>>(CA, DMAT);
        k_pair_extra<<<cdiv(NPAIR, 256), 256, 0, stream>>>(DMAT, idx, BIGA);
        gemm(BIGA, 69, P(F_E2_W), P(F_E2_B), EEMB, 32, NPAIR, 32, 69, 0);  // pair -> EEMB (32-wide)
        ln(EEMB, 32, P(F_NE2_G), P(F_NE2_B), EEMB, 32, NPAIR, 32);

        k_topk<<<LSEQ, 128, 0, stream>>>(DMAT, EIDX);
        k_feat<<<cdiv(LK, 128), 128, 0, stream>>>(H32, EEMB, CA, EIDX, BIGA, DHAT, DIST);
        gemm(BIGA, 132, P(F_M1_W), P(F_M1_B), MBUF, 64, LK, 64, 132, 1);   // silu
        gemm(MBUF, 64, P(F_M2_W), P(F_M2_B), MBUF2, 64, LK, 64, 64, 1);    // silu
        k_cat96<<<cdiv((long)LSEQ * 96, 256), 256, 0, stream>>>(H32, MBUF2, CAT96);
        lin(CAT96, 96, P(F_NU_W), P(F_NU_B), sout, 16, LSEQ, 16, 96, 0);   // h_new -> state
        lin(MBUF2, 64, P(F_VW_W), P(F_VW_B), SBUF, 3, LK, 3, 64, 0);       // s
        lin(sout, 16, P(F_VM_W), P(F_VM_B), GATE, 9, LSEQ, 9, 16, 0);      // gate
        k_update<<<cdiv(LSEQ, 128), 128, 0, stream>>>(xin, CA, SBUF, DHAT, GATE, xout);

        float* tx = xin; xin = xout; xout = tx;
        float* ts = sin_; sin_ = sout; sout = ts;
    }

    // ======================= lddt head + output =======================
    k_copy<<<cdiv(LSEQ * 9, 256), 256, 0, stream>>>(xin, out, LSEQ * 9);
    k_lddt<<<cdiv(LSEQ, 128), 128, 0, stream>>>(sin_, P(P_NSF_G), P(P_NSF_B),
                                                P(P_LD_W), P(P_LD_B), out + LSEQ * 9);
}